// GraphEncoder_44530220925002
// MI455X (gfx1250) — compile-verified
//
#include <hip/hip_runtime.h>

// CDNA5 / gfx1250: wave32, WMMA f32 16x16x4 for exact-f32 GEMM.
// Memory-bound gather (neighbor rows, ~0.9 GB @ 23.3 TB/s) + small GEMM on WMMA.

typedef float v2f __attribute__((ext_vector_type(2)));
typedef float v8f __attribute__((ext_vector_type(8)));

#define ROWS 64          // output rows per block (4 chunks of 16)
#define DIM 128          // embedding dim
#define KNBR 16          // neighbors per node
#define A_STRIDE 258     // LDS row stride in floats (2*DIM + 2 pad -> bank spread)

__global__ __launch_bounds__(256)
void GraphEncoder_sage_kernel(const int*   __restrict__ video_nodes,
                              const int*   __restrict__ neighbors,
                              const float* __restrict__ neigh_weights,
                              const float* __restrict__ emb,
                              const float* __restrict__ W1,
                              const float* __restrict__ b1,
                              float*       __restrict__ out,
                              int N)
{
    __shared__ float Atile[ROWS * A_STRIDE];   // 66048 B (<320 KB WGP LDS)

    const int tid     = threadIdx.x;
    const int rowBase = blockIdx.x * ROWS;

    // ---------------- Stage 1: build combined [ROWS x 2*DIM] tile in LDS ---
    {
        const int r    = tid >> 2;     // 0..63 : row within tile
        const int part = tid & 3;      // 0..3  : 32-dim slice
        const int d0   = part * 32;

        int row = rowBase + r;
        if (row >= N) row = N - 1;     // clamp; epilogue stores are guarded

        // normalized neighbor weights
        const float* wp = neigh_weights + (size_t)row * KNBR;
        float w[KNBR];
        float wsum = 0.f;
        #pragma unroll
        for (int k = 0; k < KNBR; ++k) { w[k] = wp[k]; wsum += w[k]; }
        const float winv = 1.0f / wsum;

        // weighted-mean gather over this thread's 32 dims
        float4 acc[8];
        #pragma unroll
        for (int j = 0; j < 8; ++j) acc[j] = make_float4(0.f, 0.f, 0.f, 0.f);

        const int* nb = neighbors + (size_t)row * KNBR;
        #pragma unroll 4
        for (int k = 0; k < KNBR; ++k) {
            const float4* ep = (const float4*)(emb + (size_t)nb[k] * DIM + d0);
            const float wk = w[k] * winv;
            #pragma unroll
            for (int j = 0; j < 8; ++j) {
                float4 e = ep[j];
                acc[j].x += wk * e.x;  acc[j].y += wk * e.y;
                acc[j].z += wk * e.z;  acc[j].w += wk * e.w;
            }
        }

        // self features
        const float4* sp = (const float4*)(emb + (size_t)video_nodes[row] * DIM + d0);

        float2* aself = (float2*)&Atile[r * A_STRIDE + d0];        // 8B-aligned
        float2* anei  = (float2*)&Atile[r * A_STRIDE + DIM + d0];
        #pragma unroll
        for (int j = 0; j < 8; ++j) {
            float4 s = sp[j];
            aself[2 * j + 0] = make_float2(s.x, s.y);
            aself[2 * j + 1] = make_float2(s.z, s.w);
            anei [2 * j + 0] = make_float2(acc[j].x, acc[j].y);
            anei [2 * j + 1] = make_float2(acc[j].z, acc[j].w);
        }
    }

    __syncthreads();

    // ---------------- Stage 2: [64x256] @ [256x128] via V_WMMA_F32_16X16X4_F32
    const int wave    = tid >> 5;      // 0..7 : output column tile
    const int lane    = tid & 31;
    const int laneLo  = lane & 15;     // N index within tile / M index of A row
    const int laneHi  = lane >> 4;     // selects K pair {0,1} vs {2,3}
    const int colBase = wave * 16;

    v8f acc0 = {}, acc1 = {}, acc2 = {}, acc3 = {};

    const float* wcol = W1 + colBase + laneLo;   // B: W1[k][colBase+laneLo]

    #pragma unroll 4
    for (int k4 = 0; k4 < 64; ++k4) {
        const int kb = k4 * 4 + laneHi * 2;

        // B fragment (4x16): VGPR0 = K=kb row, VGPR1 = K=kb+1 row (L2-resident W1)
        v2f b;
        b.x = wcol[(size_t)kb * DIM];
        b.y = wcol[(size_t)(kb + 1) * DIM];

        // A fragments (16x4) for 4 row chunks: lane holds A[laneLo][kb..kb+1]
        v2f a0 = *(const v2f*)&Atile[( 0 + laneLo) * A_STRIDE + kb];
        v2f a1 = *(const v2f*)&Atile[(16 + laneLo) * A_STRIDE + kb];
        v2f a2 = *(const v2f*)&Atile[(32 + laneLo) * A_STRIDE + kb];
        v2f a3 = *(const v2f*)&Atile[(48 + laneLo) * A_STRIDE + kb];

        acc0 = __builtin_amdgcn_wmma_f32_16x16x4_f32(false, a0, false, b, (short)0, acc0, false, false);
        acc1 = __builtin_amdgcn_wmma_f32_16x16x4_f32(false, a1, false, b, (short)0, acc1, false, false);
        acc2 = __builtin_amdgcn_wmma_f32_16x16x4_f32(false, a2, false, b, (short)0, acc2, false, false);
        acc3 = __builtin_amdgcn_wmma_f32_16x16x4_f32(false, a3, false, b, (short)0, acc3, false, false);
    }

    // ---------------- Epilogue: bias + ReLU + stores -------------------------
    const float bias = b1[colBase + laneLo];
    float* outp = out + colBase + laneLo;
    const int mBase = laneHi ? 8 : 0;             // C/D layout: VGPR j -> M=j or j+8

    if (rowBase + ROWS <= N) {
        // Fast path (all blocks except the last): straight-line stores.
        #pragma unroll
        for (int j = 0; j < 8; ++j) {
            const size_t r0 = (size_t)(rowBase + mBase + j);
            outp[(r0     ) * DIM] = fmaxf(acc0[j] + bias, 0.f);
            outp[(r0 + 16) * DIM] = fmaxf(acc1[j] + bias, 0.f);
            outp[(r0 + 32) * DIM] = fmaxf(acc2[j] + bias, 0.f);
            outp[(r0 + 48) * DIM] = fmaxf(acc3[j] + bias, 0.f);
        }
    } else {
        // Tail block: per-row guards.
        #pragma unroll
        for (int j = 0; j < 8; ++j) {
            const int r0 = rowBase + mBase + j;
            if (r0      < N) outp[(size_t)(r0     ) * DIM] = fmaxf(acc0[j] + bias, 0.f);
            if (r0 + 16 < N) outp[(size_t)(r0 + 16) * DIM] = fmaxf(acc1[j] + bias, 0.f);
            if (r0 + 32 < N) outp[(size_t)(r0 + 32) * DIM] = fmaxf(acc2[j] + bias, 0.f);
            if (r0 + 48 < N) outp[(size_t)(r0 + 48) * DIM] = fmaxf(acc3[j] + bias, 0.f);
        }
    }
}

extern "C" void kernel_launch(void* const* d_in, const int* in_sizes, int n_in,
                              void* d_out, int out_size, void* d_ws, size_t ws_size,
                              hipStream_t stream) {
    const int*   video_nodes   = (const int*)  d_in[0];   // [N]
    const int*   neighbors     = (const int*)  d_in[1];   // [N,16]
    const float* neigh_weights = (const float*)d_in[2];   // [N,16]
    const float* emb           = (const float*)d_in[3];   // [V,128]
    const float* W1            = (const float*)d_in[4];   // [256,128]
    const float* b1            = (const float*)d_in[5];   // [128]
    float*       out           = (float*)d_out;           // [N,128]

    const int N = in_sizes[0];
    const int blocks = (N + ROWS - 1) / ROWS;

    GraphEncoder_sage_kernel<<<blocks, 256, 0, stream>>>(
        video_nodes, neighbors, neigh_weights, emb, W1, b1, out, N);
}